// EarnedMemory_83013127897245
// MI455X (gfx1250) — compile-verified
//
#include <hip/hip_runtime.h>
#include <hip/hip_bf16.h>
#include <math.h>

typedef __attribute__((ext_vector_type(2))) float v2f;
typedef __attribute__((ext_vector_type(8))) float v8f;

#define MEM_M   524288
#define DIM     512
#define TOPK    32
#define NBLOCKS 512
#define ROWS_PER_BLOCK (MEM_M / NBLOCKS)          /* 1024 */
#define THREADS 256
#define NWAVES  (THREADS / 32)                    /* 8 */
#define TILES_PER_WAVE (ROWS_PER_BLOCK / (NWAVES * 16)) /* 8 */
#define NCAND   (NBLOCKS * TOPK)                  /* 16384 */
#define NEG_INF (-3.402823466e38f)

// ---------------------------------------------------------------------------
// Kernel 1: streaming scores via V_WMMA_F32_16X16X4_F32 + per-block top-32
// ---------------------------------------------------------------------------
__global__ __launch_bounds__(THREADS)
void em_score_topk(const float* __restrict__ query,
                   const float* __restrict__ keys,
                   const float* __restrict__ usefulness,
                   const float* __restrict__ access_count,
                   float* __restrict__ candScore,
                   int*   __restrict__ candIdx)
{
    __shared__ float qs[DIM];                     // query staged in LDS
    __shared__ float sc[ROWS_PER_BLOCK];          // per-row scores
    __shared__ float wscr[NWAVES * 256];          // per-wave D-matrix spill
    __shared__ float rs[THREADS];
    __shared__ int   ri[THREADS];

    const int tid  = threadIdx.x;
    const int lane = tid & 31;
    const int wid  = tid >> 5;
    const int half = lane >> 4;                   // 0 = K0/K1 lanes, 1 = K2/K3 lanes
    const int mrow = lane & 15;                   // A-matrix row within tile
    const int blockStart = blockIdx.x * ROWS_PER_BLOCK;
    const int colBase = 4 * half;

    for (int i = tid; i < DIM; i += THREADS) qs[i] = query[i];
    __syncthreads();

    for (int it = 0; it < TILES_PER_WAVE; ++it) {
        const int tile     = wid + NWAVES * it;
        const int tileBase = blockStart + tile * 16;
        const int row      = tileBase + mrow;
        const float* kp = keys + (size_t)row * DIM + colBase;
        const float* qp = qs + colBase;

        v8f acc = {};
        float ssq = 0.0f;

        // 64-column groups: stage 8 key chunks + 8 query chunks first so the
        // compiler emits a clause of 8 outstanding global_load_b128 per group
        // (memory-level parallelism >> WMMA chain depth for a BW-bound pass).
        #pragma unroll 1
        for (int k0 = 0; k0 < DIM; k0 += 64) {
            float4 a[8];
            float4 b[8];
            #pragma unroll
            for (int u = 0; u < 8; ++u)
                a[u] = *(const float4*)(kp + k0 + 8 * u);
            #pragma unroll
            for (int u = 0; u < 8; ++u)
                b[u] = *(const float4*)(qp + k0 + 8 * u);
            #pragma unroll
            for (int u = 0; u < 8; ++u) {
                v2f A1 = {a[u].x, a[u].y}, B1 = {b[u].x, b[u].y};
                v2f A2 = {a[u].z, a[u].w}, B2 = {b[u].z, b[u].w};
                acc = __builtin_amdgcn_wmma_f32_16x16x4_f32(
                          false, A1, false, B1, (short)0, acc, false, false);
                acc = __builtin_amdgcn_wmma_f32_16x16x4_f32(
                          false, A2, false, B2, (short)0, acc, false, false);
                ssq = fmaf(a[u].x, a[u].x, fmaf(a[u].y, a[u].y,
                      fmaf(a[u].z, a[u].z, fmaf(a[u].w, a[u].w, ssq))));
            }
        }

        // full ||row||^2: combine the two lane-halves of this row
        float ssqFull = ssq + __shfl_xor(ssq, 16);

        // spill accumulator: C/D layout => row m lives in VGPR (m&7), half (m>=8)
        float* wsp = &wscr[wid * 256];
        #pragma unroll
        for (int j = 0; j < 8; ++j) wsp[lane * 8 + j] = acc[j];
        __syncthreads();

        if (lane < 16) {
            const int m   = lane;
            const float dot = wsp[((m >= 8) ? 128 : 0) + (m & 7)]; // N=0 column
            const int grow = tileBase + m;
            const float w  = usefulness[grow] * __expf(-0.01f * access_count[grow]);
            const float nk = sqrtf(ssqFull);
            sc[grow - blockStart] = (dot / fmaxf(nk, 1e-12f)) * w;
        }
        __syncthreads();
    }

    // ---- per-block top-32 by iterative argmax with masking ----
    for (int r = 0; r < TOPK; ++r) {
        float best = NEG_INF;
        int   bi   = 0x7fffffff;
        for (int i = tid; i < ROWS_PER_BLOCK; i += THREADS) {
            float v = sc[i];
            if (v > best || (v == best && i < bi)) { best = v; bi = i; }
        }
        rs[tid] = best; ri[tid] = bi;
        __syncthreads();
        for (int off = THREADS / 2; off > 0; off >>= 1) {
            if (tid < off) {
                float v = rs[tid + off]; int j = ri[tid + off];
                if (v > rs[tid] || (v == rs[tid] && j < ri[tid])) { rs[tid] = v; ri[tid] = j; }
            }
            __syncthreads();
        }
        if (tid == 0) {
            int widx = ri[0];
            candScore[blockIdx.x * TOPK + r] = rs[0];
            candIdx[blockIdx.x * TOPK + r]   = blockStart + widx;
            sc[widx] = NEG_INF;
        }
        __syncthreads();
    }
}

// ---------------------------------------------------------------------------
// Kernel 2: global top-32 over candidates, then gather + mean of values rows
// ---------------------------------------------------------------------------
__global__ __launch_bounds__(512)
void em_finalize(float* __restrict__ candScore,
                 const int* __restrict__ candIdx,
                 const float* __restrict__ values,
                 float* __restrict__ out)
{
    __shared__ float rs[512];
    __shared__ int   ri[512];
    __shared__ int   sel[TOPK];
    const int tid = threadIdx.x;

    for (int r = 0; r < TOPK; ++r) {
        float best = NEG_INF;
        int   bs   = 0;
        for (int i = tid; i < NCAND; i += 512) {
            float v = candScore[i];
            if (v > best) { best = v; bs = i; }
        }
        rs[tid] = best; ri[tid] = bs;
        __syncthreads();
        for (int off = 256; off > 0; off >>= 1) {
            if (tid < off) {
                if (rs[tid + off] > rs[tid]) { rs[tid] = rs[tid + off]; ri[tid] = ri[tid + off]; }
            }
            __syncthreads();
        }
        if (tid == 0) {
            int slot = ri[0];
            sel[r] = candIdx[slot];
            candScore[slot] = NEG_INF;   // mask; rewritten by kernel 1 next call
        }
        __syncthreads();
    }

    if (tid < DIM) {
        float s = 0.0f;
        #pragma unroll
        for (int j = 0; j < TOPK; ++j)
            s += values[(size_t)sel[j] * DIM + tid];
        out[tid] = s * (1.0f / (float)TOPK);
    }
}

// ---------------------------------------------------------------------------
extern "C" void kernel_launch(void* const* d_in, const int* in_sizes, int n_in,
                              void* d_out, int out_size, void* d_ws, size_t ws_size,
                              hipStream_t stream)
{
    const float* query        = (const float*)d_in[0];
    const float* keys         = (const float*)d_in[1];
    const float* values       = (const float*)d_in[2];
    const float* usefulness   = (const float*)d_in[3];
    const float* access_count = (const float*)d_in[4];
    // d_in[5] is k == 32 (compile-time TOPK; per-block top-32 covers k<=32)

    float* candScore = (float*)d_ws;
    int*   candIdx   = (int*)((char*)d_ws + (size_t)NCAND * sizeof(float));

    em_score_topk<<<NBLOCKS, THREADS, 0, stream>>>(
        query, keys, usefulness, access_count, candScore, candIdx);
    em_finalize<<<1, 512, 0, stream>>>(
        candScore, candIdx, values, (float*)d_out);
}